// DbrxExperts_8383776161845
// MI455X (gfx1250) — compile-verified
//
#include <hip/hip_runtime.h>
#include <hip/hip_bf16.h>

typedef float v2f __attribute__((ext_vector_type(2)));
typedef float v8f __attribute__((ext_vector_type(8)));

#define NE   16      // experts
#define TOPK 4
#define HID  2048
#define FFN  4096
#define NT   64      // tokens (B*S)
#define KC   32      // K chunk staged through LDS
#define XST  36      // padded LDS row stride (floats): conflict-free & 16B aligned

// ---------- CDNA5 async global->LDS path ----------
// Probe-revealed signature: (int4 addrspace(1)*, int4 addrspace(3)*, imm offset, imm cpol)
#if __has_builtin(__builtin_amdgcn_global_load_async_to_lds_b128)
#define ASYNC_LDS 1
typedef int v4i_vec __attribute__((vector_size(4 * sizeof(int))));
typedef __attribute__((address_space(1))) v4i_vec* g_v4i_ptr;
typedef __attribute__((address_space(3))) v4i_vec* l_v4i_ptr;
#else
#define ASYNC_LDS 0
#endif

__device__ __forceinline__ void copy16_to_lds(const float* g, float* l) {
#if ASYNC_LDS
    __builtin_amdgcn_global_load_async_to_lds_b128(
        (g_v4i_ptr)g, (l_v4i_ptr)l, 0 /*imm offset*/, 0 /*cpol*/);
#else
    *(float4*)l = *(const float4*)g;    // sync fallback: global_load_b128 + ds_store_b128
#endif
}

__device__ __forceinline__ void async_copy_wait() {
#if ASYNC_LDS
#if __has_builtin(__builtin_amdgcn_s_wait_asynccnt)
    __builtin_amdgcn_s_wait_asynccnt(0);
#else
    asm volatile("s_wait_asynccnt 0x0" ::: "memory");
#endif
#endif
}

// ---------------- Phase 0: zero output ----------------
__global__ void zero_kernel(float* out, int n) {
    int i = blockIdx.x * blockDim.x + threadIdx.x;
    if (i < n) out[i] = 0.0f;
}

// ---------------- Phase 1: routing (compact per-expert token lists) ----------------
__global__ void route_kernel(const int* __restrict__ top_experts,
                             const float* __restrict__ top_weights,
                             int* __restrict__ tok_list,
                             float* __restrict__ w_list,
                             int* __restrict__ counts) {
    __shared__ float r[NT][NE];
    int t = threadIdx.x;              // 64 threads
    if (t < NT) {
        for (int e = 0; e < NE; ++e) r[t][e] = 0.0f;
        for (int k = 0; k < TOPK; ++k) {
            int e = top_experts[t * TOPK + k] & (NE - 1);
            r[t][e] += top_weights[t * TOPK + k];   // duplicates accumulate, like .at[].add
        }
    }
    __syncthreads();
    if (t < NE) {
        int cnt = 0;
        for (int tok = 0; tok < NT; ++tok) {
            float v = r[tok][t];
            if (v != 0.0f) { tok_list[t * NT + cnt] = tok; w_list[t * NT + cnt] = v; ++cnt; }
        }
        counts[t] = cnt;
        for (int i = cnt; i < NT; ++i) { tok_list[t * NT + i] = 0; w_list[t * NT + i] = 0.0f; }
    }
}

// ---------------- Phase 2: gate/up GEMMs + silu + routing scale ----------------
// grid (FFN/64, NE), 256 threads (8 wave32). Wave w: M-tile = w&3, N-tiles {2*(w>>2), 2*(w>>2)+1}.
// Double-buffered LDS: async-load chunk k+1 while WMMA-consuming chunk k.
__global__ void __launch_bounds__(256)
gateup_kernel(const float* __restrict__ x, const float* __restrict__ w1,
              const float* __restrict__ v1, const int* __restrict__ tok_list,
              const float* __restrict__ w_list, const int* __restrict__ counts,
              float* __restrict__ inter) {
    const int e  = blockIdx.y;
    const int f0 = blockIdx.x * 64;

    __shared__ float xs [2][NT * XST];
    __shared__ float w1s[2][64 * XST];
    __shared__ float v1s[2][64 * XST];
    __shared__ int   stok[NT];
    __shared__ float swt[NT];
    __shared__ int   sactive;

    const int tid  = threadIdx.x;
    const int lane = tid & 31;
    const int wave = tid >> 5;

    if (tid < NT) { stok[tid] = tok_list[e * NT + tid]; swt[tid] = w_list[e * NT + tid]; }
    if (tid == 0) sactive = (counts[e] + 15) >> 4;
    __syncthreads();
    const int active = sactive;
    if (active == 0) return;                    // uniform across block

    const int  m_tile = wave & 3;
    const int  n2     = wave >> 2;
    const bool do_compute = (m_tile < active);  // wave-uniform: EXEC all-1s inside

    v8f accg0 = {}, accg1 = {}, accu0 = {}, accu1 = {};

    const float* w1e = w1 + (size_t)(e * FFN + f0) * HID;
    const float* v1e = v1 + (size_t)(e * FFN + f0) * HID;

    // stage one K-chunk into LDS buffer `buf` (each thread: 6 x b128)
    auto stage = [&](int buf, int k0) {
#pragma unroll
        for (int j = 0; j < 2; ++j) {
            int idx = tid + j * 256;            // 0..511
            int row = idx >> 3;                 // 0..63
            int c4  = (idx & 7) * 4;            // 0..28
            copy16_to_lds(x   + (size_t)stok[row] * HID + k0 + c4, &xs [buf][row * XST + c4]);
            copy16_to_lds(w1e + (size_t)row       * HID + k0 + c4, &w1s[buf][row * XST + c4]);
            copy16_to_lds(v1e + (size_t)row       * HID + k0 + c4, &v1s[buf][row * XST + c4]);
        }
    };

    stage(0, 0);
    async_copy_wait();
    __syncthreads();

    for (int k0 = 0; k0 < HID; k0 += KC) {
        const int cur = (k0 / KC) & 1;
        if (k0 + KC < HID) stage(cur ^ 1, k0 + KC);   // prefetch next chunk (async)

        if (do_compute) {
            const float* xb = xs [cur];
            const float* wb = w1s[cur];
            const float* vb = v1s[cur];
            const int am    = m_tile * 16 + (lane & 15);
            const int bn0   = (n2 * 2 + 0) * 16 + (lane & 15);
            const int bn1   = (n2 * 2 + 1) * 16 + (lane & 15);
            const int khalf = (lane >> 4) << 1;          // A/B frag K split across half-waves
#pragma unroll
            for (int kk = 0; kk < KC; kk += 4) {
                const int kb = kk + khalf;
                v2f a;  a.x  = xb[am  * XST + kb]; a.y  = xb[am  * XST + kb + 1];
                v2f b0; b0.x = wb[bn0 * XST + kb]; b0.y = wb[bn0 * XST + kb + 1];
                v2f b1; b1.x = wb[bn1 * XST + kb]; b1.y = wb[bn1 * XST + kb + 1];
                v2f c0; c0.x = vb[bn0 * XST + kb]; c0.y = vb[bn0 * XST + kb + 1];
                v2f c1; c1.x = vb[bn1 * XST + kb]; c1.y = vb[bn1 * XST + kb + 1];
                accg0 = __builtin_amdgcn_wmma_f32_16x16x4_f32(false, a, false, b0, (short)0, accg0, false, false);
                accg1 = __builtin_amdgcn_wmma_f32_16x16x4_f32(false, a, false, b1, (short)0, accg1, false, false);
                accu0 = __builtin_amdgcn_wmma_f32_16x16x4_f32(false, a, false, c0, (short)0, accu0, false, false);
                accu1 = __builtin_amdgcn_wmma_f32_16x16x4_f32(false, a, false, c1, (short)0, accu1, false, false);
            }
        }
        async_copy_wait();     // next-chunk LDS writes landed
        __syncthreads();       // all waves done reading `cur`, next buffer published
    }

    if (do_compute) {
        const int col   = lane & 15;
        const int rhalf = (lane >> 4) << 3;     // C/D frag: lanes 16-31 hold rows M+8
#pragma unroll
        for (int i = 0; i < 8; ++i) {
            int   trow = m_tile * 16 + i + rhalf;
            float wt   = swt[trow];             // zero for padded slots -> inter == 0
            float g0 = accg0[i], u0 = accu0[i];
            float g1 = accg1[i], u1 = accu1[i];
            float s0 = (g0 / (1.0f + __expf(-g0))) * u0 * wt;   // silu(gate)*up*route
            float s1 = (g1 / (1.0f + __expf(-g1))) * u1 * wt;
            size_t base = ((size_t)e * NT + trow) * FFN + f0;
            inter[base + (n2 * 2 + 0) * 16 + col] = s0;
            inter[base + (n2 * 2 + 1) * 16 + col] = s1;
        }
    }
}

// ---------------- Phase 3: down GEMM + scatter-accumulate into out ----------------
// grid (HID/64, NE), 256 threads. K = FFN. Same double-buffered async pipeline.
__global__ void __launch_bounds__(256)
down_kernel(const float* __restrict__ w2, const float* __restrict__ inter,
            const int* __restrict__ tok_list, const int* __restrict__ counts,
            float* __restrict__ out) {
    const int e  = blockIdx.y;
    const int h0 = blockIdx.x * 64;

    __shared__ float is [2][NT * XST];     // [64 tok][KC f]
    __shared__ float w2s[2][KC * 64];      // [KC f][64 h]
    __shared__ int   stok[NT];
    __shared__ int   sactive;

    const int tid  = threadIdx.x;
    const int lane = tid & 31;
    const int wave = tid >> 5;

    if (tid < NT) stok[tid] = tok_list[e * NT + tid];
    if (tid == 0) sactive = (counts[e] + 15) >> 4;
    __syncthreads();
    const int active = sactive;
    if (active == 0) return;

    const int  m_tile = wave & 3;
    const int  n2     = wave >> 2;
    const bool do_compute = (m_tile < active);

    v8f acc0 = {}, acc1 = {};

    const float* w2e = w2    + (size_t)e * FFN * HID;
    const float* ie  = inter + (size_t)e * NT  * FFN;

    auto stage = [&](int buf, int k0) {
#pragma unroll
        for (int j = 0; j < 2; ++j) {
            int idx = tid + j * 256;
            {   // intermediate tile: 64 rows x 8 float4
                int row = idx >> 3, c4 = (idx & 7) * 4;
                copy16_to_lds(ie + (size_t)row * FFN + k0 + c4, &is[buf][row * XST + c4]);
            }
            {   // w2 tile: 32 rows x 16 float4
                int row = idx >> 4, c4 = (idx & 15) * 4;
                copy16_to_lds(w2e + (size_t)(k0 + row) * HID + h0 + c4, &w2s[buf][row * 64 + c4]);
            }
        }
    };

    stage(0, 0);
    async_copy_wait();
    __syncthreads();

    for (int k0 = 0; k0 < FFN; k0 += KC) {
        const int cur = (k0 / KC) & 1;
        if (k0 + KC < FFN) stage(cur ^ 1, k0 + KC);

        if (do_compute) {
            const float* ib = is [cur];
            const float* wb = w2s[cur];
            const int am    = m_tile * 16 + (lane & 15);
            const int nn0   = (n2 * 2 + 0) * 16 + (lane & 15);
            const int nn1   = (n2 * 2 + 1) * 16 + (lane & 15);
            const int khalf = (lane >> 4) << 1;
#pragma unroll
            for (int kk = 0; kk < KC; kk += 4) {
                const int kb = kk + khalf;
                v2f a;  a.x  = ib[am * XST + kb];      a.y  = ib[am * XST + kb + 1];
                v2f b0; b0.x = wb[kb * 64 + nn0];      b0.y = wb[(kb + 1) * 64 + nn0];
                v2f b1; b1.x = wb[kb * 64 + nn1];      b1.y = wb[(kb + 1) * 64 + nn1];
                acc0 = __builtin_amdgcn_wmma_f32_16x16x4_f32(false, a, false, b0, (short)0, acc0, false, false);
                acc1 = __builtin_amdgcn_wmma_f32_16x16x4_f32(false, a, false, b1, (short)0, acc1, false, false);
            }
        }
        async_copy_wait();
        __syncthreads();
    }

    if (do_compute) {
        const int col   = lane & 15;
        const int rhalf = (lane >> 4) << 3;
#pragma unroll
        for (int i = 0; i < 8; ++i) {
            int trow = m_tile * 16 + i + rhalf;
            int tok  = stok[trow];              // padded rows -> token 0, value 0.0 (harmless)
            atomicAdd(out + (size_t)tok * HID + h0 + (n2 * 2 + 0) * 16 + col, acc0[i]);
            atomicAdd(out + (size_t)tok * HID + h0 + (n2 * 2 + 1) * 16 + col, acc1[i]);
        }
    }
}

// ---------------- launch ----------------
extern "C" void kernel_launch(void* const* d_in, const int* in_sizes, int n_in,
                              void* d_out, int out_size, void* d_ws, size_t ws_size,
                              hipStream_t stream) {
    const float* x           = (const float*)d_in[0];
    // d_in[1] = full router softmax "weights": unused by the reference math
    const float* top_weights = (const float*)d_in[2];
    const int*   top_experts = (const int*)d_in[3];
    const float* w1          = (const float*)d_in[4];
    const float* v1          = (const float*)d_in[5];
    const float* w2          = (const float*)d_in[6];
    float*       out         = (float*)d_out;

    char* ws = (char*)d_ws;
    const size_t inter_bytes = (size_t)NE * NT * FFN * sizeof(float);   // 16 MB
    float* inter    = (float*)ws;
    int*   tok_list = (int*)  (ws + inter_bytes);
    float* w_list   = (float*)(ws + inter_bytes + (size_t)NE * NT * sizeof(int));
    int*   counts   = (int*)  (ws + inter_bytes + (size_t)NE * NT * (sizeof(int) + sizeof(float)));

    zero_kernel<<<(out_size + 255) / 256, 256, 0, stream>>>(out, out_size);
    route_kernel<<<1, 64, 0, stream>>>(top_experts, top_weights, tok_list, w_list, counts);

    dim3 g1(FFN / 64, NE);
    gateup_kernel<<<g1, 256, 0, stream>>>(x, w1, v1, tok_list, w_list, counts, inter);

    dim3 g2(HID / 64, NE);
    down_kernel<<<g2, 256, 0, stream>>>(w2, inter, tok_list, counts, out);
}